// LSTM_76605036692127
// MI455X (gfx1250) — compile-verified
//
#include <hip/hip_runtime.h>
#include <math.h>

// ---------------------------------------------------------------------------
// LSTM: T=512, B=64, D=1024, H=1024
//   Kernel A: Z = X @ Wi + b           (parallel over all T*B rows)
//   Kernel B: one per time step, z = Z[t] + h@Wh, fused gates + c/h update
// fp32 throughout, V_WMMA_F32_16X16X4_F32 for the matmuls.
// Kernel B streams Wh tiles L2 -> LDS with the Tensor Data Mover (TDM),
// double-buffered on TENSORcnt.
// ---------------------------------------------------------------------------

typedef __attribute__((ext_vector_type(2))) float v2f;
typedef __attribute__((ext_vector_type(8))) float v8f;
typedef __attribute__((ext_vector_type(4))) unsigned int u32x4;
typedef __attribute__((ext_vector_type(8))) int i32x8;
typedef __attribute__((ext_vector_type(4))) int i32x4;

static constexpr int T  = 512;
static constexpr int Bc = 64;     // batch
static constexpr int D  = 1024;
static constexpr int H  = 1024;
static constexpr int NG = 4 * H;  // 4096 fused gate width

// LDS pitches (floats), chosen for bank-conflict-free WMMA fragment reads.
static constexpr int SA_P  = 68;   // A tile 64 x 64: column reads hit banks 4m+k
static constexpr int SBP_P = 72;   // B tile 64 x 64 (pregemm), 2*72 % 64 = 16
static constexpr int SBS_P = 136;  // B tile 64 x 128 (step),   2*136 % 64 = 16

// ---------------------------------------------------------------------------
// Kernel A: Z[m, n] = sum_k X[m, k] * Wi[k, n] + b[n]
//   M = T*B = 32768, K = 1024, N = 4096.  64x64 tile per workgroup.
// ---------------------------------------------------------------------------
__global__ __launch_bounds__(256) void lstm_pregemm(
    const float* __restrict__ X,    // [T*B, D]
    const float* __restrict__ Wi,   // [D, NG]
    const float* __restrict__ bias, // [NG]
    float* __restrict__ Z)          // [T*B, NG]
{
  __shared__ float sA[64 * SA_P];
  __shared__ float sB[64 * SBP_P];

  const int tid  = threadIdx.x;
  const int lane = tid & 31;
  const int wave = tid >> 5;
  const int half = lane >> 4;      // selects K pair for WMMA fragments
  const int l16  = lane & 15;

  const int row0 = blockIdx.y * 64;
  const int col0 = blockIdx.x * 64;

  // Wave -> two 16x16 C tiles: (mt, ntb) and (mt, ntb+1); A frag reused.
  const int mt  = wave & 3;
  const int ntb = (wave >> 2) * 2;

  v8f acc0 = {};
  v8f acc1 = {};

  for (int kc = 0; kc < D; kc += 64) {
    {
      const int r  = tid >> 4;
      const int c4 = (tid & 15) * 4;
#pragma unroll
      for (int i = 0; i < 4; ++i) {
        float4 va = *(const float4*)(X + (size_t)(row0 + r + i * 16) * D + kc + c4);
        *(float4*)(&sA[(r + i * 16) * SA_P + c4]) = va;
        float4 vb = *(const float4*)(Wi + (size_t)(kc + r + i * 16) * NG + col0 + c4);
        *(float4*)(&sB[(r + i * 16) * SBP_P + c4]) = vb;
      }
    }
    __syncthreads();

#pragma unroll
    for (int kk = 0; kk < 64; kk += 4) {
      const int kb = kk + half * 2;
      v2f a, b0, b1;
      a.x  = sA[(mt * 16 + l16) * SA_P + kb];
      a.y  = sA[(mt * 16 + l16) * SA_P + kb + 1];
      b0.x = sB[kb * SBP_P + ntb * 16 + l16];
      b0.y = sB[(kb + 1) * SBP_P + ntb * 16 + l16];
      b1.x = sB[kb * SBP_P + (ntb + 1) * 16 + l16];
      b1.y = sB[(kb + 1) * SBP_P + (ntb + 1) * 16 + l16];
      acc0 = __builtin_amdgcn_wmma_f32_16x16x4_f32(false, a, false, b0,
                                                   (short)0, acc0, false, false);
      acc1 = __builtin_amdgcn_wmma_f32_16x16x4_f32(false, a, false, b1,
                                                   (short)0, acc1, false, false);
    }
    __syncthreads();
  }

  // C/D layout: VGPR r, lane L: M = 16*mt + r + 8*(L/16), N = 16*nt + L%16.
#pragma unroll
  for (int r = 0; r < 8; ++r) {
    const int m  = mt * 16 + r + half * 8;
    const int n0 = col0 + ntb * 16 + l16;
    const int n1 = n0 + 16;
    Z[(size_t)(row0 + m) * NG + n0] = acc0[r] + bias[n0];
    Z[(size_t)(row0 + m) * NG + n1] = acc1[r] + bias[n1];
  }
}

// ---------------------------------------------------------------------------
// TDM descriptor: load one Wh chunk as a 3D tile into LDS.
//   X: 32 contiguous cols (one gate strip)            -> tile_dim0 = 32
//   Y: 4 gate strips, stride 1024 elements            -> tile_dim1 = 4
//   Z: 64 K-rows, stride 4096 elements                -> tile_dim2 = 64
// LDS receives rows of 128 floats; pad 8 DWORDs every 128 DWORDs
// (pad_interval code 6 = 128 DW, pad_amount code 7 = 8 DW) -> pitch 136.
// ---------------------------------------------------------------------------
__device__ __forceinline__ void tdm_load_wh_tile(const float* gbase,
                                                 unsigned lds_byte_addr) {
  const unsigned long long ga = (unsigned long long)(uintptr_t)gbase;

  u32x4 g0;
  g0.x = 1u;                                   // count=1, user descriptor
  g0.y = lds_byte_addr;                        // lds_addr
  g0.z = (unsigned)(ga & 0xFFFFFFFFu);         // global_addr[31:0]
  g0.w = (unsigned)((ga >> 32) & 0x01FFFFFFu)  // global_addr[56:32]
         | (2u << 30);                         // type = 2 ("image")

  i32x8 g1;
  g1[0] = (2 << 16)      // data_size = 4 bytes
        | (1 << 20)      // pad_enable
        | (6 << 22)      // pad_interval: 128 DWORDs
        | (7 << 25);     // pad_amount:   8 DWORDs
  g1[1] = (int)(4096u << 16);        // tensor_dim0 = 4096 (bits 79:48 lo)
  g1[2] = (int)(4u << 16);           // tensor_dim0 hi = 0 ; tensor_dim1 = 4
  g1[3] = (int)(32u << 16);          // tensor_dim1 hi = 0 ; tile_dim0 = 32
  g1[4] = (int)(4u | (64u << 16));   // tile_dim1 = 4 ; tile_dim2 = 64
  g1[5] = 1024;                      // tensor_dim0_stride lo32 (Y stride)
  g1[6] = (int)(4096u << 16);        // dim0_stride hi = 0 ; dim1_stride lo16
  g1[7] = 0;                         // tensor_dim1_stride hi32 (4096 >> 16)

  i32x4 g2;
  g2[0] = 1024;                      // tensor_dim2 (>= tile_dim2, OOB bound)
  g2[1] = 0;                         // tensor_dim3
  g2[2] = 0;                         // tensor_dim2_stride lo32 (unused, 3D)
  g2[3] = 0;                         // dim2_stride hi ; tile_dim3 = 0

  i32x4 g3 = {0, 0, 0, 0};
  i32x8 g4 = {0, 0, 0, 0, 0, 0, 0, 0};  // 6-arg builtin: unused trailing group

  __builtin_amdgcn_tensor_load_to_lds(g0, g1, g2, g3, g4, 0);
}

// ---------------------------------------------------------------------------
// Kernel B: one time step.  grid.x = H/32 = 32 workgroups.
//   Wave w: gate g = w>>1, 16-column sub-strip ns = (w&1)*16, full M=64.
//   Wh tiles are DMA'd by the TDM (wave 0 issues, TENSORcnt double buffer).
// ---------------------------------------------------------------------------
__global__ __launch_bounds__(256) void lstm_step(
    const float* __restrict__ Zt,    // [B, NG] pre-activations for step t
    const float* __restrict__ hprev, // [B, H]
    float* __restrict__ c,           // [B, H] carried cell state (ws)
    float* __restrict__ ht,          // [B, H] output ys[t]
    const float* __restrict__ Wh)    // [H, NG]
{
  __shared__ float sA[64 * SA_P];    // h tile    64(B) x 64(K)
  __shared__ float sB0[64 * SBS_P];  // Wh tile   64(K) x 128 (double buffer 0)
  __shared__ float sB1[64 * SBS_P];  // Wh tile   (double buffer 1)
  float* zs = sB0;                   // aliased after K loop: [4][64][33]

  const int tid  = threadIdx.x;
  const int lane = tid & 31;
  const int wave = tid >> 5;
  const int half = lane >> 4;
  const int l16  = lane & 15;

  const int n0 = blockIdx.x * 32;    // H-column base of this WG
  const int g  = wave >> 1;          // gate handled by this wave
  const int ns = (wave & 1) * 16;    // sub-strip within the 32 columns

  // Generic LDS addresses truncate to the LDS byte offset (aperture layout).
  const unsigned offB0 = (unsigned)(uintptr_t)(void*)sB0;
  const unsigned offB1 = (unsigned)(uintptr_t)(void*)sB1;
  const float* whb = Wh + n0;        // gate-0 strip base column

  v8f acc[4];
#pragma unroll
  for (int mt = 0; mt < 4; ++mt) acc[mt] = (v8f){};

  // Prime the pipeline: chunk 0 -> sB0.
  if (wave == 0) tdm_load_wh_tile(whb, offB0);

  for (int kci = 0; kci < H / 64; ++kci) {
    const int kc = kci * 64;
    // sA <- hprev[0:64, kc:kc+64] (regular coalesced path; only 16 KB)
    {
      const int r  = tid >> 4;
      const int c4 = (tid & 15) * 4;
#pragma unroll
      for (int i = 0; i < 4; ++i) {
        float4 v = *(const float4*)(hprev + (size_t)(r + i * 16) * H + kc + c4);
        *(float4*)(&sA[(r + i * 16) * SA_P + c4]) = v;
      }
    }
    // Wave 0: kick next chunk's DMA, then wait for the current one.
    // Tensor ops from one wave complete in order, so TENSORcnt <= 1 after
    // issuing chunk k+1 guarantees chunk k has landed in LDS.
    if (wave == 0) {
      if (kci + 1 < H / 64) {
        tdm_load_wh_tile(whb + (size_t)(kc + 64) * NG,
                         ((kci + 1) & 1) ? offB1 : offB0);
        __builtin_amdgcn_s_wait_tensorcnt(1);
      } else {
        __builtin_amdgcn_s_wait_tensorcnt(0);
      }
    }
    __syncthreads();

    const float* sB = (kci & 1) ? sB1 : sB0;
#pragma unroll
    for (int kk = 0; kk < 64; kk += 4) {
      const int kb   = kk + half * 2;
      const int bcol = g * 32 + ns + l16;
      v2f b;
      b.x = sB[kb * SBS_P + bcol];
      b.y = sB[(kb + 1) * SBS_P + bcol];
#pragma unroll
      for (int mt = 0; mt < 4; ++mt) {
        v2f a;
        a.x = sA[(mt * 16 + l16) * SA_P + kb];
        a.y = sA[(mt * 16 + l16) * SA_P + kb + 1];
        acc[mt] = __builtin_amdgcn_wmma_f32_16x16x4_f32(false, a, false, b,
                                                        (short)0, acc[mt],
                                                        false, false);
      }
    }
    __syncthreads();  // fences tile reads before next DMA / zs alias writes
  }

  // z = h@Wh + Z[t]; stash per-gate strips in LDS for cross-wave fusion.
#pragma unroll
  for (int mt = 0; mt < 4; ++mt) {
#pragma unroll
    for (int r = 0; r < 8; ++r) {
      const int m    = mt * 16 + r + half * 8;
      const int ncol = ns + l16;              // 0..31 inside gate tile
      const int gcol = g * H + n0 + ncol;     // global fused-gate column
      const float z  = acc[mt][r] + Zt[(size_t)m * NG + gcol];
      zs[g * (64 * 33) + m * 33 + ncol] = z;
    }
  }
  __syncthreads();

  // Fused gate nonlinearity + state update: 64*32 = 2048 outputs, 8/thread.
#pragma unroll
  for (int j = 0; j < 8; ++j) {
    const int idx = tid + j * 256;
    const int m   = idx >> 5;
    const int n   = idx & 31;
    const float zi = zs[0 * 2112 + m * 33 + n];
    const float zf = zs[1 * 2112 + m * 33 + n];
    const float zg = zs[2 * 2112 + m * 33 + n];
    const float zo = zs[3 * 2112 + m * 33 + n];

    const float ig = 1.0f / (1.0f + __expf(-zi));
    const float fg = 1.0f / (1.0f + __expf(-zf));
    const float gg = tanhf(zg);
    const float og = 1.0f / (1.0f + __expf(-zo));

    const int hidx = m * H + n0 + n;
    const float cold = c[hidx];
    const float cnew = fg * cold + ig * gg;
    const float hnew = og * tanhf(cnew);
    c[hidx]  = cnew;
    ht[hidx] = hnew;
  }
}

// ---------------------------------------------------------------------------
// Launch: inputs {x, c0, h0, Wi, Wh, b}; output = concat(ys, cT, hT).
// Workspace: Z [T*B, 4H] fp32 (512 MiB) + c carry [B, H] (256 KiB).
// ---------------------------------------------------------------------------
extern "C" void kernel_launch(void* const* d_in, const int* in_sizes, int n_in,
                              void* d_out, int out_size, void* d_ws, size_t ws_size,
                              hipStream_t stream) {
  const float* x  = (const float*)d_in[0];
  const float* c0 = (const float*)d_in[1];
  const float* h0 = (const float*)d_in[2];
  const float* Wi = (const float*)d_in[3];
  const float* Wh = (const float*)d_in[4];
  const float* b  = (const float*)d_in[5];

  float* ys = (float*)d_out;                       // [T, B, H]
  float* cT = ys + (size_t)T * Bc * H;             // [B, H]
  float* hT = cT + (size_t)Bc * H;                 // [B, H]

  float* Z   = (float*)d_ws;                       // [T*B, NG]
  float* cws = Z + (size_t)T * Bc * NG;            // [B, H]

  // c <- c0 (carried state for the scan)
  (void)hipMemcpyAsync(cws, c0, (size_t)Bc * H * sizeof(float),
                       hipMemcpyDeviceToDevice, stream);

  // Kernel A: all-timestep input projection (massively parallel).
  dim3 gridA(NG / 64, (T * Bc) / 64);              // 64 x 512 workgroups
  lstm_pregemm<<<gridA, 256, 0, stream>>>(x, Wi, b, Z);

  // Kernel B: sequential recurrence, one launch per step.
  for (int t = 0; t < T; ++t) {
    const float* hp = (t == 0) ? h0 : (ys + (size_t)(t - 1) * Bc * H);
    lstm_step<<<dim3(H / 32), 256, 0, stream>>>(
        Z + (size_t)t * Bc * NG, hp, cws, ys + (size_t)t * Bc * H, Wh);
  }

  // Final states.
  (void)hipMemcpyAsync(cT, cws, (size_t)Bc * H * sizeof(float),
                       hipMemcpyDeviceToDevice, stream);
  (void)hipMemcpyAsync(hT, ys + (size_t)(T - 1) * Bc * H,
                       (size_t)Bc * H * sizeof(float),
                       hipMemcpyDeviceToDevice, stream);
}